// EnhancedGraphEncoder_30502857736298
// MI455X (gfx1250) — compile-verified
//
#include <hip/hip_runtime.h>

// ---------------------------------------------------------------------------
// EnhancedGraphEncoder forward for MI455X (gfx1250, wave32).
//  - GEMMs: V_WMMA_F32_16X16X4_F32 (exact fp32 matrix path), W staged into
//    LDS once per workgroup via GLOBAL_LOAD_ASYNC_TO_LDS_B128 (ASYNCcnt).
//  - Aggregation/BN/pool: global_atomic_add_f32 (unsafeAtomicAdd); the whole
//    per-layer working set (~165MB) is L2-resident (192MB L2).
// ---------------------------------------------------------------------------

#define NN 100000
#define NE 1600000
#define NG 128

typedef __attribute__((ext_vector_type(2))) float v2f;
typedef __attribute__((ext_vector_type(8))) float v8f;
typedef int v4i __attribute__((__vector_size__(16)));  // matches builtin param

#define AS_GLOBAL __attribute__((address_space(1)))
#define AS_LDS __attribute__((address_space(3)))

#if __has_builtin(__builtin_amdgcn_global_load_async_to_lds_b128)
#define HAVE_ASYNC_LDS 1
#else
#define HAVE_ASYNC_LDS 0
#endif

// ---------------- degree / norm precompute ----------------

__global__ __launch_bounds__(256) void deg_kernel(const int* __restrict__ dst,
                                                  float* __restrict__ deg, int ne) {
  int e = blockIdx.x * blockDim.x + threadIdx.x;
  if (e < ne) unsafeAtomicAdd(&deg[dst[e]], 1.0f);
}

__global__ __launch_bounds__(256) void dinv_kernel(float* __restrict__ deg, int n) {
  int i = blockIdx.x * blockDim.x + threadIdx.x;
  if (i < n) deg[i] = rsqrtf(deg[i] + 1.0f);  // +1 self loop => always >= 1
}

__global__ __launch_bounds__(256) void norm_kernel(const int* __restrict__ src,
                                                   const int* __restrict__ dst,
                                                   const float* __restrict__ dinv,
                                                   float* __restrict__ nrm, int ne) {
  int e = blockIdx.x * blockDim.x + threadIdx.x;
  if (e < ne) nrm[e] = dinv[src[e]] * dinv[dst[e]];
}

// ---------------- GEMM: Out[N x FOUT] = A[N x 128] @ W[128 x FOUT] ----------
// One wave per 16-row tile; FOUT/16 v8f accumulators; K in steps of 4 via
// V_WMMA_F32_16X16X4_F32.  W is staged in LDS (row stride FOUT+16 so the
// +2-row access of lanes 16..31 lands at bank offset +32: conflict-free).
// A frag: lane L -> A[L&15][k0+(L>>4)*2 +{0,1}].
// B frag: lane L -> W[k0+(L>>4)*2 +{0,1}][tile*16 + (L&15)].
// D: vgpr r -> row r+8*(L>>4), col L&15.

template <int FOUT>
__global__ __launch_bounds__(256) void gemm_wmma(const float* __restrict__ A,
                                                 const float* __restrict__ W,
                                                 float* __restrict__ Out, int ntiles) {
  constexpr int K = 128;
  constexpr int NT = FOUT / 16;
  constexpr int SSTRIDE = FOUT + 16;  // pad: +2 rows => +32 banks

  __shared__ float sW[K * SSTRIDE];

  // Cooperative stage of W[128 x FOUT] into LDS (all 256 threads, 16B chunks).
#pragma unroll
  for (int i = threadIdx.x * 4; i < K * FOUT; i += 256 * 4) {
    const int r = i / FOUT;
    const int c = i % FOUT;
#if HAVE_ASYNC_LDS
    __builtin_amdgcn_global_load_async_to_lds_b128(
        (AS_GLOBAL v4i*)(W + i), (AS_LDS v4i*)(&sW[r * SSTRIDE + c]), 0, 0);
#else
    *(float4*)(&sW[r * SSTRIDE + c]) = *(const float4*)(W + i);
#endif
  }
#if HAVE_ASYNC_LDS
#if __has_builtin(__builtin_amdgcn_s_wait_asynccnt)
  __builtin_amdgcn_s_wait_asynccnt(0);
#else
  asm volatile("s_wait_asynccnt 0x0" ::: "memory");
#endif
#endif
  __syncthreads();

  const int wave = (int)((blockIdx.x * blockDim.x + threadIdx.x) >> 5);
  if (wave < ntiles) {  // wave-uniform: EXEC stays all-ones inside
    const int lane = (int)(threadIdx.x & 31);
    const int m = lane & 15;
    const int hb = lane >> 4;

    const float* arow = A + (size_t)(wave * 16 + m) * K + hb * 2;
    v8f acc[NT] = {};

#pragma unroll 4
    for (int k0 = 0; k0 < K; k0 += 4) {
      const float2 av = *(const float2*)(arow + k0);
      v2f a;
      a[0] = av.x;
      a[1] = av.y;
      const int boff = (k0 + hb * 2) * SSTRIDE + m;
#pragma unroll
      for (int t = 0; t < NT; ++t) {
        v2f b;
        b[0] = sW[boff + t * 16];
        b[1] = sW[boff + SSTRIDE + t * 16];
        acc[t] = __builtin_amdgcn_wmma_f32_16x16x4_f32(false, a, false, b, (short)0,
                                                       acc[t], false, false);
      }
    }

    float* orow = Out + (size_t)(wave * 16 + 8 * hb) * FOUT + m;
#pragma unroll
    for (int t = 0; t < NT; ++t) {
#pragma unroll
      for (int r = 0; r < 8; ++r) orow[(size_t)r * FOUT + t * 16] = acc[t][r];
    }
  }
}

// ---------------- edge aggregation: Agg[dst] += H[src] * norm ---------------
// One wave per edge (incl. self loops appended after real edges).

template <int F>
__global__ __launch_bounds__(256) void edge_agg(const float* __restrict__ H,
                                                const int* __restrict__ src,
                                                const int* __restrict__ dst,
                                                const float* __restrict__ nrmE,
                                                const float* __restrict__ dinv,
                                                float* __restrict__ Agg,
                                                int nedges, int nnodes) {
  const long long wave = ((long long)blockIdx.x * blockDim.x + threadIdx.x) >> 5;
  const int lane = (int)(threadIdx.x & 31);
  const long long total = (long long)nedges + nnodes;
  if (wave >= total) return;
  int s, d;
  float nm;
  if (wave < nedges) {
    s = src[wave];
    d = dst[wave];
    nm = nrmE[wave];
  } else {  // self loop
    s = d = (int)(wave - nedges);
    float di = dinv[s];
    nm = di * di;
  }
  constexpr int PER = F / 32;
  const float* hp = H + (size_t)s * F + lane * PER;
  float* ap = Agg + (size_t)d * F + lane * PER;
  if constexpr (PER == 4) {
    float4 v = *(const float4*)hp;
    unsafeAtomicAdd(ap + 0, v.x * nm);
    unsafeAtomicAdd(ap + 1, v.y * nm);
    unsafeAtomicAdd(ap + 2, v.z * nm);
    unsafeAtomicAdd(ap + 3, v.w * nm);
  } else {
    float2 v = *(const float2*)hp;
    unsafeAtomicAdd(ap + 0, v.x * nm);
    unsafeAtomicAdd(ap + 1, v.y * nm);
  }
}

// ---------------- BatchNorm: column stats + apply ---------------------------
// Bias b cancels exactly inside BN (mean-shift invariance), so it is skipped.

template <int F>
__global__ void bn_stats(const float* __restrict__ H, float* __restrict__ sum,
                         float* __restrict__ sq, int n, int rpb) {
  int c = threadIdx.x;  // blockDim.x == F
  int r0 = blockIdx.x * rpb;
  int r1 = r0 + rpb;
  if (r1 > n) r1 = n;
  float s = 0.f, q = 0.f;
  for (int r = r0; r < r1; ++r) {
    float v = H[(size_t)r * F + c];
    s += v;
    q += v * v;
  }
  unsafeAtomicAdd(&sum[c], s);
  unsafeAtomicAdd(&sq[c], q);
}

template <int F, bool RELU, bool SKIP>
__global__ __launch_bounds__(256) void bn_apply(float* __restrict__ H,
                                                const float* __restrict__ sum,
                                                const float* __restrict__ sq,
                                                const float* __restrict__ gamma,
                                                const float* __restrict__ beta,
                                                const float* __restrict__ Hprev,
                                                const float* __restrict__ skipw, int n) {
  long long idx = (long long)blockIdx.x * blockDim.x + threadIdx.x;
  long long total = (long long)n * F;
  if (idx >= total) return;
  int c = (int)(idx & (F - 1));
  float invn = 1.0f / (float)n;
  float mu = sum[c] * invn;
  float var = fmaxf(sq[c] * invn - mu * mu, 0.0f);
  float y = gamma[c] * (H[idx] - mu) * rsqrtf(var + 1e-5f) + beta[c];
  if (RELU) y = fmaxf(y, 0.0f);
  if (SKIP) y = fmaf(skipw[0], Hprev[idx], y);
  H[idx] = y;
}

// ---------------- global mean pool ------------------------------------------

__global__ __launch_bounds__(256) void pool_cnt(const int* __restrict__ batch,
                                                float* __restrict__ cnt, int n) {
  int i = blockIdx.x * blockDim.x + threadIdx.x;
  if (i < n) unsafeAtomicAdd(&cnt[batch[i]], 1.0f);
}

__global__ __launch_bounds__(256) void pool_sum(const float* __restrict__ H,
                                                const int* __restrict__ batch,
                                                float* __restrict__ sums, int n) {
  int t = blockIdx.x * blockDim.x + threadIdx.x;  // 16 threads per row
  int row = t >> 4;
  if (row >= n) return;
  int c4 = (t & 15) * 4;
  int g = batch[row];
  const float4 v = *(const float4*)(H + (size_t)row * 64 + c4);
  float* sp = sums + (size_t)g * 64 + c4;
  unsafeAtomicAdd(sp + 0, v.x);
  unsafeAtomicAdd(sp + 1, v.y);
  unsafeAtomicAdd(sp + 2, v.z);
  unsafeAtomicAdd(sp + 3, v.w);
}

__global__ __launch_bounds__(256) void pool_div(const float* __restrict__ sums,
                                                const float* __restrict__ cnt,
                                                float* __restrict__ out) {
  int i = blockIdx.x * blockDim.x + threadIdx.x;
  if (i < NG * 64) out[i] = sums[i] / fmaxf(cnt[i >> 6], 1.0f);
}

// ---------------------------------------------------------------------------

extern "C" void kernel_launch(void* const* d_in, const int* in_sizes, int n_in,
                              void* d_out, int out_size, void* d_ws, size_t ws_size,
                              hipStream_t stream) {
  (void)in_sizes; (void)n_in; (void)out_size; (void)ws_size;
  const float* x     = (const float*)d_in[0];
  const int*   ei    = (const int*)d_in[1];
  const int*   batch = (const int*)d_in[2];
  const float* W0 = (const float*)d_in[3];
  const float* g0 = (const float*)d_in[5];
  const float* be0 = (const float*)d_in[6];
  const float* W1 = (const float*)d_in[7];
  const float* g1 = (const float*)d_in[9];
  const float* be1 = (const float*)d_in[10];
  const float* W2 = (const float*)d_in[11];
  const float* g2 = (const float*)d_in[13];
  const float* be2 = (const float*)d_in[14];
  const float* W3 = (const float*)d_in[15];
  const float* g3 = (const float*)d_in[17];
  const float* be3 = (const float*)d_in[18];
  const float* skipw = (const float*)d_in[19];
  float* out = (float*)d_out;

  const int N = NN, E = NE;
  const int* src = ei;
  const int* dst = ei + E;

  // workspace layout (floats)
  float* ws   = (float*)d_ws;
  float* deg  = ws;                         // N (becomes dinv after dinv_kernel)
  float* nrmE = deg + N;                    // E
  float* hA   = nrmE + E;                   // N*128
  float* hB   = hA + (size_t)N * 128;       // N*128
  float* hC   = hB + (size_t)N * 128;       // N*128
  float* bns  = hC + (size_t)N * 128;       // 128
  float* bnq  = bns + 128;                  // 128
  float* psum = bnq + 128;                  // 128*64
  float* pcnt = psum + 128 * 64;            // 128

  // ---- degrees + symmetric norms ----
  (void)hipMemsetAsync(deg, 0, (size_t)N * sizeof(float), stream);
  deg_kernel<<<(E + 255) / 256, 256, 0, stream>>>(dst, deg, E);
  dinv_kernel<<<(N + 255) / 256, 256, 0, stream>>>(deg, N);
  norm_kernel<<<(E + 255) / 256, 256, 0, stream>>>(src, dst, deg, nrmE, E);

  const int ntiles = N / 16;                                 // 6250 (exact)
  const int gemm_blocks = (ntiles * 32 + 255) / 256;
  const int agg_blocks = (int)((((long long)(E + N)) * 32 + 255) / 256);
  const int stats_blocks = (N + 255) / 256;
  const int ew128_blocks = (int)(((long long)N * 128 + 255) / 256);
  const int ew64_blocks = (int)(((long long)N * 64 + 255) / 256);

  // ---- layer 0: h = relu(BN(GCN(x, W0))) -> hC ----
  gemm_wmma<128><<<gemm_blocks, 256, 0, stream>>>(x, W0, hB, ntiles);
  (void)hipMemsetAsync(hC, 0, (size_t)N * 128 * sizeof(float), stream);
  edge_agg<128><<<agg_blocks, 256, 0, stream>>>(hB, src, dst, nrmE, deg, hC, E, N);
  (void)hipMemsetAsync(bns, 0, 128 * sizeof(float), stream);
  (void)hipMemsetAsync(bnq, 0, 128 * sizeof(float), stream);
  bn_stats<128><<<stats_blocks, 128, 0, stream>>>(hC, bns, bnq, N, 256);
  bn_apply<128, true, false><<<ew128_blocks, 256, 0, stream>>>(hC, bns, bnq, g0, be0,
                                                               nullptr, skipw, N);

  // ---- layer 1: h = relu(BN(GCN(hC, W1))) + sw*hC -> hB ----
  gemm_wmma<128><<<gemm_blocks, 256, 0, stream>>>(hC, W1, hA, ntiles);
  (void)hipMemsetAsync(hB, 0, (size_t)N * 128 * sizeof(float), stream);
  edge_agg<128><<<agg_blocks, 256, 0, stream>>>(hA, src, dst, nrmE, deg, hB, E, N);
  (void)hipMemsetAsync(bns, 0, 128 * sizeof(float), stream);
  (void)hipMemsetAsync(bnq, 0, 128 * sizeof(float), stream);
  bn_stats<128><<<stats_blocks, 128, 0, stream>>>(hB, bns, bnq, N, 256);
  bn_apply<128, true, true><<<ew128_blocks, 256, 0, stream>>>(hB, bns, bnq, g1, be1,
                                                              hC, skipw, N);

  // ---- layer 2: h = relu(BN(GCN(hB, W2))) + sw*hB -> hC ----
  gemm_wmma<128><<<gemm_blocks, 256, 0, stream>>>(hB, W2, hA, ntiles);
  (void)hipMemsetAsync(hC, 0, (size_t)N * 128 * sizeof(float), stream);
  edge_agg<128><<<agg_blocks, 256, 0, stream>>>(hA, src, dst, nrmE, deg, hC, E, N);
  (void)hipMemsetAsync(bns, 0, 128 * sizeof(float), stream);
  (void)hipMemsetAsync(bnq, 0, 128 * sizeof(float), stream);
  bn_stats<128><<<stats_blocks, 128, 0, stream>>>(hC, bns, bnq, N, 256);
  bn_apply<128, true, true><<<ew128_blocks, 256, 0, stream>>>(hC, bns, bnq, g2, be2,
                                                              hB, skipw, N);

  // ---- layer 3 (F=64, no relu/skip): h = BN(GCN(hC, W3)) -> hB ----
  gemm_wmma<64><<<gemm_blocks, 256, 0, stream>>>(hC, W3, hA, ntiles);
  (void)hipMemsetAsync(hB, 0, (size_t)N * 64 * sizeof(float), stream);
  edge_agg<64><<<agg_blocks, 256, 0, stream>>>(hA, src, dst, nrmE, deg, hB, E, N);
  (void)hipMemsetAsync(bns, 0, 64 * sizeof(float), stream);
  (void)hipMemsetAsync(bnq, 0, 64 * sizeof(float), stream);
  bn_stats<64><<<stats_blocks, 64, 0, stream>>>(hB, bns, bnq, N, 256);
  bn_apply<64, false, false><<<ew64_blocks, 256, 0, stream>>>(hB, bns, bnq, g3, be3,
                                                              nullptr, skipw, N);

  // ---- global mean pool ----
  (void)hipMemsetAsync(psum, 0, 128 * 64 * sizeof(float), stream);
  (void)hipMemsetAsync(pcnt, 0, 128 * sizeof(float), stream);
  pool_cnt<<<(N + 255) / 256, 256, 0, stream>>>(batch, pcnt, N);
  pool_sum<<<(N * 16 + 255) / 256, 256, 0, stream>>>(hB, batch, psum, N);
  pool_div<<<(NG * 64 + 255) / 256, 256, 0, stream>>>(psum, pcnt, out);
}